// GCN_34454227649229
// MI455X (gfx1250) — compile-verified
//
#include <hip/hip_runtime.h>

// Problem constants (match the reference).
#define N_NODES 100000
#define N_EDGES 1600000
#define IN_C    128
#define HID_C   64
#define OUT_C   32

typedef float v2f __attribute__((ext_vector_type(2)));
typedef float v8f __attribute__((ext_vector_type(8)));

// ---------------------------------------------------------------------------
// Degree / normalization: deg[i] = 1 (self loop) + #edges into i; dinv = rsqrt.
// ---------------------------------------------------------------------------
__global__ void gcn_init_deg(float* __restrict__ deg) {
    int i = blockIdx.x * blockDim.x + threadIdx.x;
    if (i < N_NODES) deg[i] = 1.0f;   // self-loop contribution
}

__global__ void gcn_accum_deg(const int* __restrict__ dst, float* __restrict__ deg) {
    int e = blockIdx.x * blockDim.x + threadIdx.x;
    if (e < N_EDGES) atomicAdd(&deg[dst[e]], 1.0f);
}

__global__ void gcn_dinv(float* __restrict__ deg) {
    int i = blockIdx.x * blockDim.x + threadIdx.x;
    if (i < N_NODES) deg[i] = rsqrtf(deg[i]);   // deg >= 1 always (self loops)
}

// ---------------------------------------------------------------------------
// FP32 WMMA GEMM:  H[M,COUT] = (RELU ? relu(A) : A)[M,K] @ W[K,COUT]
// One wave -> one 16-row stripe, all COUT columns (COUT/16 accumulators).
// Uses V_WMMA_F32_16X16X4_F32 (exact fp32, matches the fp32 reference).
//
// A-frag (16x4 f32, 2 VGPRs/lane):  row = lane&15, ks = k0 + 2*(lane>>4)
// B-frag (4x16 f32, 2 VGPRs/lane):  col = lane&15, same ks
// C/D   (16x16 f32, 8 VGPRs/lane):  M = r + 8*(lane>>4), N = lane&15
// ---------------------------------------------------------------------------
template <int K, int COUT, bool RELU>
__global__ void gcn_gemm_wmma_f32(const float* __restrict__ A,
                                  const float* __restrict__ W,
                                  float* __restrict__ H,
                                  int M) {
    const int lane  = threadIdx.x & 31;
    const int wave  = threadIdx.x >> 5;
    const int tile  = blockIdx.x * (blockDim.x >> 5) + wave;   // wave-uniform
    if (tile * 16 >= M) return;                                // whole-wave exit

    const int m0    = tile * 16;
    const int r15   = lane & 15;          // A row / B col
    const int khalf = (lane >> 4) << 1;   // 0 or 2

    constexpr int NT = COUT / 16;
    v8f acc[NT] = {};

    for (int k0 = 0; k0 < K; k0 += 4) {
        const int kk = k0 + khalf;
        // A fragment (contiguous pair along K within a row)
        const float* ap = A + (size_t)(m0 + r15) * K + kk;
        v2f a;
        a.x = ap[0];
        a.y = ap[1];
        if (RELU) {
            a.x = fmaxf(a.x, 0.0f);
            a.y = fmaxf(a.y, 0.0f);
        }
#pragma unroll
        for (int j = 0; j < NT; ++j) {
            const int n = j * 16 + r15;
            v2f b;
            b.x = W[(size_t)kk * COUT + n];
            b.y = W[(size_t)(kk + 1) * COUT + n];
            acc[j] = __builtin_amdgcn_wmma_f32_16x16x4_f32(
                /*neg_a=*/false, a, /*neg_b=*/false, b,
                /*c_mod=*/(short)0, acc[j],
                /*reuse_a=*/false, /*reuse_b=*/false);
        }
    }

    // Store D: VGPR r -> row m0 + r + 8*(lane>>4), col = j*16 + (lane&15)
    const int mo = (lane >> 4) << 3;
#pragma unroll
    for (int j = 0; j < NT; ++j) {
        const int n = j * 16 + r15;
#pragma unroll
        for (int r = 0; r < 8; ++r) {
            H[(size_t)(m0 + mo + r) * COUT + n] = acc[j][r];
        }
    }
}

// ---------------------------------------------------------------------------
// Aggregation init: out[i,c] = bias[c] + dinv[i]^2 * h[i,c]
// (self-loop term folded in, so the scatter only walks the 1.6M real edges)
// ---------------------------------------------------------------------------
template <int C>
__global__ void gcn_init_agg(float* __restrict__ out,
                             const float* __restrict__ h,
                             const float* __restrict__ bias,
                             const float* __restrict__ dinv) {
    long long t = (long long)blockIdx.x * blockDim.x + threadIdx.x;
    if (t >= (long long)N_NODES * C) return;
    const int i = (int)(t / C);
    const int c = (int)(t % C);
    const float di = dinv[i];
    out[t] = bias[c] + di * di * h[t];
}

// ---------------------------------------------------------------------------
// Edge scatter: out[dst,c] += dinv[src]*dinv[dst] * h[src,c]
// Thread layout: consecutive threads = consecutive channels of one edge, so
// the h[src] gather is a coalesced C*4-byte burst; atomics are f32 L2 RMWs.
// ---------------------------------------------------------------------------
template <int C>
__global__ void gcn_scatter(const int* __restrict__ src,
                            const int* __restrict__ dst,
                            const float* __restrict__ dinv,
                            const float* __restrict__ h,
                            float* __restrict__ out) {
    long long t = (long long)blockIdx.x * blockDim.x + threadIdx.x;
    if (t >= (long long)N_EDGES * C) return;
    const int e = (int)(t / C);
    const int c = (int)(t % C);
    const int s = src[e];
    const int d = dst[e];
    const float norm = dinv[s] * dinv[d];
    atomicAdd(out + (size_t)d * C + c, norm * h[(size_t)s * C + c]);
}

// ---------------------------------------------------------------------------
// Launch
// ---------------------------------------------------------------------------
extern "C" void kernel_launch(void* const* d_in, const int* in_sizes, int n_in,
                              void* d_out, int out_size, void* d_ws, size_t ws_size,
                              hipStream_t stream) {
    const float* x   = (const float*)d_in[0];           // [N, 128]
    const int*   ei  = (const int*)d_in[1];             // [2, E] (int32)
    const float* W1  = (const float*)d_in[2];           // [128, 64]
    const float* b1  = (const float*)d_in[3];           // [64]
    const float* W2  = (const float*)d_in[4];           // [64, 32]
    const float* b2  = (const float*)d_in[5];           // [32]
    const int*   src = ei;
    const int*   dst = ei + N_EDGES;
    float*       out = (float*)d_out;                   // [N, 32]

    // Workspace layout (floats)
    float* ws    = (float*)d_ws;
    float* dinv  = ws;                                      // 100000 (pad 102400)
    float* h1    = ws + 102400;                             // N * 64
    float* agg1  = h1 + (size_t)N_NODES * HID_C;            // N * 64
    float* h2    = agg1 + (size_t)N_NODES * HID_C;          // N * 32
    (void)in_sizes; (void)n_in; (void)out_size; (void)ws_size;

    const int B = 256;

    // 1) degrees -> dinv
    gcn_init_deg<<<(N_NODES + B - 1) / B, B, 0, stream>>>(dinv);
    gcn_accum_deg<<<(N_EDGES + B - 1) / B, B, 0, stream>>>(dst, dinv);
    gcn_dinv<<<(N_NODES + B - 1) / B, B, 0, stream>>>(dinv);

    // 2) layer 1: h1 = x @ W1 (WMMA fp32), agg1 = b1 + selfloop + scatter
    const int TILES = N_NODES / 16;            // 6250, exact
    const int WPB   = B / 32;                  // 8 waves per block
    gcn_gemm_wmma_f32<IN_C, HID_C, false>
        <<<(TILES + WPB - 1) / WPB, B, 0, stream>>>(x, W1, h1, N_NODES);
    {
        long long tot = (long long)N_NODES * HID_C;
        gcn_init_agg<HID_C><<<(int)((tot + B - 1) / B), B, 0, stream>>>(agg1, h1, b1, dinv);
        long long etot = (long long)N_EDGES * HID_C;
        gcn_scatter<HID_C><<<(int)((etot + B - 1) / B), B, 0, stream>>>(src, dst, dinv, h1, agg1);
    }

    // 3) layer 2: h2 = relu(agg1) @ W2 (WMMA fp32, ReLU fused in A-load)
    gcn_gemm_wmma_f32<HID_C, OUT_C, true>
        <<<(TILES + WPB - 1) / WPB, B, 0, stream>>>(agg1, W2, h2, N_NODES);
    {
        long long tot = (long long)N_NODES * OUT_C;
        gcn_init_agg<OUT_C><<<(int)((tot + B - 1) / B), B, 0, stream>>>(out, h2, b2, dinv);
        long long etot = (long long)N_EDGES * OUT_C;
        gcn_scatter<OUT_C><<<(int)((etot + B - 1) / B), B, 0, stream>>>(src, dst, dinv, h2, out);
    }
}